// CharRNN_22290880266385
// MI455X (gfx1250) — compile-verified
//
#include <hip/hip_runtime.h>

// ---------------------------------------------------------------------------
// CharRNN forward for MI455X (gfx1250, wave32, WMMA).
//
//   h_{t+1} = tanh( projb[tok_t] + h_t @ Wh )     projb[v] = emb[v]@Wx + b
//   logits  = (h_T * drop_mask) @ Wd + bd
//
// Per wave: 16 batch rows. h@Wh done with V_WMMA_F32_16X16X4_F32 (K=10 padded
// to 12 -> 3 WMMA chunks). projb gather lands directly in the C/D accumulator
// layout so WMMA produces the full pre-activation. D->A relayout goes through
// a per-wave LDS tile (stride 20 floats: conflict-free for both access
// patterns), ordered with s_wait_dscnt (same-wave LDS is in-order).
// ---------------------------------------------------------------------------

typedef float v2f __attribute__((ext_vector_type(2)));
typedef float v8f __attribute__((ext_vector_type(8)));

#define B_TOT  16384
#define T_LEN  100
#define V_SZ   256
#define E_SZ   50
#define H_SZ   10
#define L_SZ   15
#define HP     16      // padded hidden width of projb table
#define ROWS_PER_BLOCK 128
#define TPB    256     // 8 waves of 32
#define T_CHUNK 20     // token staging chunk (divides T_LEN)
#define H_STRIDE 20    // padded LDS row stride for h tile (bank-conflict free)

__device__ __forceinline__ float fast_tanh(float x) {
#if __has_builtin(__builtin_amdgcn_tanhf)
    return __builtin_amdgcn_tanhf(x);           // v_tanh_f32 (CDNA5 trans op)
#else
    // tanh(x) = (e^{2x}-1)/(e^{2x}+1), e^{2x} = 2^{2*log2(e)*x}
    x = fminf(fmaxf(x, -9.0f), 9.0f);           // keep exp2 finite
    float z = __builtin_amdgcn_exp2f(x * 2.885390081777927f);
    return (z - 1.0f) * __builtin_amdgcn_rcpf(z + 1.0f);
#endif
}

__device__ __forceinline__ v8f wmma_f32_16x16x4(v2f a, v2f b, v8f c) {
    return __builtin_amdgcn_wmma_f32_16x16x4_f32(
        /*neg_a=*/false, a, /*neg_b=*/false, b,
        /*c_mod=*/(short)0, c, /*reuse_a=*/false, /*reuse_b=*/false);
}

__global__ __launch_bounds__(TPB)
void charrnn_kernel(const int*   __restrict__ x,      // [B, T]
                    const float* __restrict__ emb,    // [V, E]
                    const float* __restrict__ Wx,     // [E, H]
                    const float* __restrict__ Wh,     // [H, H]
                    const float* __restrict__ b_rnn,  // [H]
                    const float* __restrict__ Wd,     // [H, L]
                    const float* __restrict__ bd,     // [L]
                    const float* __restrict__ dmask,  // [B, H]
                    float*       __restrict__ out)    // [B, L]
{
    __shared__ float s_projb[V_SZ * HP];                    // 16 KB
    __shared__ int   s_tok[ROWS_PER_BLOCK * T_CHUNK];       // 10 KB
    __shared__ float s_h[(TPB / 32) * 16 * H_STRIDE];       // 10 KB

    const int tid   = threadIdx.x;
    const int lane  = tid & 31;
    const int wave  = tid >> 5;
    const int n     = lane & 15;       // N column (hidden unit / logit unit)
    const int half  = lane >> 4;       // lane half (0/1)
    const int mBase = half * 8;        // D-layout: vgpr r holds row m = r + mBase
    const int rowLocal = wave * 16;    // wave's base row within block
    const int gRowBase = blockIdx.x * ROWS_PER_BLOCK + rowLocal;

    // ---- build projb[v][0..15] = emb[v]@Wx + b  (one row per thread) ----
    {
        const int v = tid;             // TPB == V_SZ == 256
        float s[H_SZ];
#pragma unroll
        for (int j = 0; j < H_SZ; ++j) s[j] = b_rnn[j];
        for (int e = 0; e < E_SZ; ++e) {
            const float em = emb[v * E_SZ + e];
#pragma unroll
            for (int j = 0; j < H_SZ; ++j)
                s[j] = fmaf(em, Wx[e * H_SZ + j], s[j]);
        }
#pragma unroll
        for (int j = 0; j < HP; ++j)
            s_projb[v * HP + j] = (j < H_SZ) ? s[j] : 0.0f;
    }

    // ---- Wh as WMMA B operands: 3 chunks of 4x16, vgpr j: K = 4c+j+2*half ----
    v2f whb[3];
#pragma unroll
    for (int c = 0; c < 3; ++c) {
#pragma unroll
        for (int j = 0; j < 2; ++j) {
            const int K = 4 * c + j + 2 * half;
            whb[c][j] = (K < H_SZ && n < H_SZ) ? Wh[K * H_SZ + n] : 0.0f;
        }
    }

    const float colMask = (n < H_SZ) ? 1.0f : 0.0f;
    float* hw = &s_h[wave * 16 * H_STRIDE];     // per-wave 16x(20) h tile

    // h0 = 0
#pragma unroll
    for (int r = 0; r < 8; ++r) hw[(r + mBase) * H_STRIDE + n] = 0.0f;

    __syncthreads();   // projb staged, h tile zeroed

    // ---------------------------- recurrence ----------------------------
    for (int tc = 0; tc < T_LEN; tc += T_CHUNK) {
        __syncthreads();   // previous chunk fully consumed
        for (int i = tid; i < ROWS_PER_BLOCK * T_CHUNK; i += TPB) {
            const int rl = i / T_CHUNK, tt = i % T_CHUNK;
            s_tok[i] = x[(blockIdx.x * ROWS_PER_BLOCK + rl) * T_LEN + tc + tt];
        }
        __syncthreads();

        for (int t = 0; t < T_CHUNK; ++t) {
            // A = h_prev in 16x4-chunk layout (ISA 7.12.2): lane m = n,
            // vgpr j -> K = 4c + j + 2*half. Even float offset -> b64 loads.
            v2f a[3];
#pragma unroll
            for (int c = 0; c < 3; ++c) {
                const float* p = &hw[n * H_STRIDE + 4 * c + 2 * half];
                a[c][0] = p[0];
                a[c][1] = p[1];
            }

            // C = projb[token] gathered straight into D layout:
            // vgpr r holds row m = r + mBase (token uniform per lane-half
            // -> LDS broadcast), column n.
            v8f acc;
#pragma unroll
            for (int r = 0; r < 8; ++r) {
                const int tok = s_tok[(rowLocal + r + mBase) * T_CHUNK + t];
                acc[r] = s_projb[tok * HP + n];
            }

            acc = wmma_f32_16x16x4(a[0], whb[0], acc);
            acc = wmma_f32_16x16x4(a[1], whb[1], acc);
            acc = wmma_f32_16x16x4(a[2], whb[2], acc);

            // h = tanh(pre), zero padded columns, write back for next A
#pragma unroll
            for (int r = 0; r < 8; ++r)
                hw[(r + mBase) * H_STRIDE + n] = fast_tanh(acc[r]) * colMask;

            // cross-lane LDS RAW: next step's A loads read this wave's stores
            asm volatile("s_wait_dscnt 0" ::: "memory");
        }
    }

    // ------------------- dropout + dense layer (WMMA) -------------------
    // A = h_T * drop_mask (applied during A load), B = Wd, C = bias tile.
    v2f af[3], wdb[3];
#pragma unroll
    for (int c = 0; c < 3; ++c) {
#pragma unroll
        for (int j = 0; j < 2; ++j) {
            const int K = 4 * c + j + 2 * half;   // hidden index
            float hv = 0.0f;
            if (K < H_SZ)
                hv = hw[n * H_STRIDE + K] * dmask[(gRowBase + n) * H_SZ + K];
            af[c][j]  = hv;
            wdb[c][j] = (K < H_SZ && n < L_SZ) ? Wd[K * L_SZ + n] : 0.0f;
        }
    }

    const float bdv = (n < L_SZ) ? bd[n] : 0.0f;
    v8f lac;
#pragma unroll
    for (int r = 0; r < 8; ++r) lac[r] = bdv;

    lac = wmma_f32_16x16x4(af[0], wdb[0], lac);
    lac = wmma_f32_16x16x4(af[1], wdb[1], lac);
    lac = wmma_f32_16x16x4(af[2], wdb[2], lac);

    if (n < L_SZ) {
#pragma unroll
        for (int r = 0; r < 8; ++r)
            out[(gRowBase + r + mBase) * L_SZ + n] = lac[r];
    }
}

extern "C" void kernel_launch(void* const* d_in, const int* in_sizes, int n_in,
                              void* d_out, int out_size, void* d_ws, size_t ws_size,
                              hipStream_t stream) {
    (void)in_sizes; (void)n_in; (void)d_ws; (void)ws_size; (void)out_size;
    const int*   x     = (const int*)  d_in[0];
    const float* emb   = (const float*)d_in[1];
    const float* Wx    = (const float*)d_in[2];
    const float* Wh    = (const float*)d_in[3];
    const float* b_rnn = (const float*)d_in[4];
    const float* Wd    = (const float*)d_in[5];
    const float* bd    = (const float*)d_in[6];
    const float* dmask = (const float*)d_in[7];
    float* out = (float*)d_out;

    charrnn_kernel<<<B_TOT / ROWS_PER_BLOCK, TPB, 0, stream>>>(
        x, emb, Wx, Wh, b_rnn, Wd, bd, dmask, out);
}